// MoELayer_35948876267751
// MI455X (gfx1250) — compile-verified
//
#include <hip/hip_runtime.h>
#include <math.h>

// Problem dims (fixed by the reference)
#define G    8
#define S    2048
#define MDIM 2048
#define H    8192
#define E    8
#define SC   16   // sequence chunks for deterministic dispatch reduction

typedef __attribute__((ext_vector_type(2))) float v2f;
typedef __attribute__((ext_vector_type(8))) float v8f;

// ---------------------------------------------------------------------------
// Kernel 1: gating scores -> softmax(E) -> top2 -> renorm -> cw[G*S][E]
// One wave per token; lanes stride over M.
// ---------------------------------------------------------------------------
__global__ __launch_bounds__(256) void k_gating(const float* __restrict__ x,
                                                const float* __restrict__ Wg,
                                                const float* __restrict__ bg,
                                                float* __restrict__ cw)
{
    int tid  = threadIdx.x;
    int lane = tid & 31;
    int wave = tid >> 5;
    int token = blockIdx.x * 8 + wave;           // G*S tokens total
    const float* xrow = x + (size_t)token * MDIM;

    float acc[E];
#pragma unroll
    for (int e = 0; e < E; ++e) acc[e] = 0.f;

    for (int m = lane; m < MDIM; m += 32) {
        float xv = xrow[m];
        const float* wrow = Wg + (size_t)m * E;  // 8 consecutive floats
#pragma unroll
        for (int e = 0; e < E; ++e) acc[e] += xv * wrow[e];
    }
    // wave32 cross-lane reduction
#pragma unroll
    for (int e = 0; e < E; ++e) {
#pragma unroll
        for (int off = 16; off > 0; off >>= 1)
            acc[e] += __shfl_xor(acc[e], off, 32);
    }
    if (lane == 0) {
        float l[E];
        float mx = -1e30f;
#pragma unroll
        for (int e = 0; e < E; ++e) { l[e] = acc[e] + bg[e]; mx = fmaxf(mx, l[e]); }
        float sum = 0.f;
#pragma unroll
        for (int e = 0; e < E; ++e) { l[e] = expf(l[e] - mx); sum += l[e]; }
        float inv = 1.f / sum;
#pragma unroll
        for (int e = 0; e < E; ++e) l[e] *= inv;   // softmax over E
        // top-2 (first-index tie break, matching top_k)
        int i0 = 0; float p0 = l[0];
#pragma unroll
        for (int e = 1; e < E; ++e) if (l[e] > p0) { p0 = l[e]; i0 = e; }
        int i1 = (i0 == 0) ? 1 : 0; float p1 = l[i1];
#pragma unroll
        for (int e = 0; e < E; ++e) if (e != i0 && l[e] > p1) { p1 = l[e]; i1 = e; }
        // softmax over the two selected probabilities
        float t  = expf(p1 - p0);
        float w0 = 1.f / (1.f + t);
        float w1 = t  / (1.f + t);
        float* cwr = cw + (size_t)token * E;
#pragma unroll
        for (int e = 0; e < E; ++e) cwr[e] = 0.f;
        cwr[i0] = w0; cwr[i1] = w1;
    }
}

// ---------------------------------------------------------------------------
// Kernel 2a: dispatch pooling partials: dvp[sc][e][g][m] = sum over s-chunk of
//            (cw[g,s,e] > 0) * x[g,s,m]   (deterministic, no atomics)
// ---------------------------------------------------------------------------
__global__ __launch_bounds__(256) void k_dispatch_part(const float* __restrict__ x,
                                                       const float* __restrict__ cw,
                                                       float* __restrict__ dvp)
{
    int tid = threadIdx.x;
    int b = blockIdx.x;
    int sc = b % SC;            b /= SC;
    int mc = b % (MDIM / 256);  b /= (MDIM / 256);
    int g  = b;
    int m  = mc * 256 + tid;

    float acc[E];
#pragma unroll
    for (int e = 0; e < E; ++e) acc[e] = 0.f;

    int s0 = sc * (S / SC);
    for (int s = s0; s < s0 + (S / SC); ++s) {
        float xv = x[((size_t)g * S + s) * MDIM + m];
        const float* cwr = cw + ((size_t)g * S + s) * E;
#pragma unroll
        for (int e = 0; e < E; ++e) acc[e] += (cwr[e] > 0.f) ? xv : 0.f;
    }
#pragma unroll
    for (int e = 0; e < E; ++e)
        dvp[(((size_t)sc * E + e) * G + g) * MDIM + m] = acc[e];
}

// Kernel 2b: fixed-order reduce over SC chunks -> dvec[e][g][m]
__global__ __launch_bounds__(256) void k_dispatch_reduce(const float* __restrict__ dvp,
                                                         float* __restrict__ dvec)
{
    size_t i = (size_t)blockIdx.x * 256 + threadIdx.x;  // over E*G*MDIM
    float a = 0.f;
#pragma unroll
    for (int sc = 0; sc < SC; ++sc) a += dvp[(size_t)sc * (E * G * MDIM) + i];
    dvec[i] = a;
}

// ---------------------------------------------------------------------------
// Kernel 3: h[e] = silu( wi[e] (HxM) @ dvec[e] (Mx16) ), fp32 WMMA 16x16x4.
// One wave per 16-row H tile. B columns: 0..7 = groups, 8..15 = zero.
// A layout (32-bit 16x4): lanes 0-15 rows, VGPR0/1 = K0,K1; lanes 16-31 = K2,K3.
// ---------------------------------------------------------------------------
__global__ __launch_bounds__(256) void k_expert_in(const float* __restrict__ wi,
                                                   const float* __restrict__ dvec,
                                                   float* __restrict__ hbuf)
{
    int tid  = threadIdx.x;
    int lane = tid & 31;
    int wv   = (blockIdx.x << 3) + (tid >> 5);   // E * (H/16) wave-tiles
    int e    = wv / (H / 16);
    int ht   = wv % (H / 16);
    int row  = ht * 16 + (lane & 15);
    int khalf = (lane >> 4) * 2;
    int n    = lane & 15;

    const float* Abase = wi   + ((size_t)e * H + row) * MDIM + khalf;
    const float* Bbase = dvec + ((size_t)e * G + (n & 7)) * MDIM + khalf;
    v2f zero = 0.0f;
    bool bvalid = (n < 8);

    v8f c = {};
    for (int k0 = 0; k0 < MDIM; k0 += 4) {
        v2f a  = *(const v2f*)(Abase + k0);
        v2f bt = *(const v2f*)(Bbase + k0);
        v2f b  = bvalid ? bt : zero;
        c = __builtin_amdgcn_wmma_f32_16x16x4_f32(false, a, false, b,
                                                  (short)0, c, false, false);
        if ((k0 & 255) == 0)
            __builtin_prefetch((const void*)(Abase + k0 + 512), 0, 1);
    }
    // D layout: lanes 0-15 VGPRj -> row j; lanes 16-31 VGPRj -> row 8+j
    int rbase = ht * 16 + ((lane >> 4) << 3);
#pragma unroll
    for (int j = 0; j < 8; ++j) {
        float v  = c[j];
        float sv = v / (1.f + expf(-v));         // silu
        hbuf[((size_t)e * H + rbase + j) * 16 + n] = sv;
    }
}

// ---------------------------------------------------------------------------
// Kernel 4: eo[g][e][m] = wo[e] (MxH) @ h[e] (Hx16), fp32 WMMA 16x16x4.
// ---------------------------------------------------------------------------
__global__ __launch_bounds__(256) void k_expert_out(const float* __restrict__ wo,
                                                    const float* __restrict__ hbuf,
                                                    float* __restrict__ eob)
{
    int tid  = threadIdx.x;
    int lane = tid & 31;
    int wv   = (blockIdx.x << 3) + (tid >> 5);   // E * (MDIM/16) wave-tiles
    int e    = wv / (MDIM / 16);
    int mt   = wv % (MDIM / 16);
    int row  = mt * 16 + (lane & 15);
    int khalf = (lane >> 4) * 2;
    int n    = lane & 15;

    const float* Abase = wo   + ((size_t)e * MDIM + row) * H + khalf;
    const float* Bbase = hbuf + (size_t)e * H * 16 + (size_t)khalf * 16 + n;

    v8f c = {};
    for (int k0 = 0; k0 < H; k0 += 4) {
        v2f a = *(const v2f*)(Abase + k0);
        v2f b;
        b.x = Bbase[(size_t)k0 * 16];
        b.y = Bbase[(size_t)k0 * 16 + 16];
        c = __builtin_amdgcn_wmma_f32_16x16x4_f32(false, a, false, b,
                                                  (short)0, c, false, false);
        if ((k0 & 255) == 0)
            __builtin_prefetch((const void*)(Abase + k0 + 512), 0, 1);
    }
    int rbase = mt * 16 + ((lane >> 4) << 3);
    if (n < 8) {                                   // columns 8..15 are padding
#pragma unroll
        for (int j = 0; j < 8; ++j)
            eob[((size_t)n * E + e) * MDIM + rbase + j] = c[j];
    }
}

// ---------------------------------------------------------------------------
// Kernel 5: out[g,s,m] = sum_e 2*cw[g,s,e] * eo[g][e][m]
// ---------------------------------------------------------------------------
__global__ __launch_bounds__(256) void k_combine(const float* __restrict__ cw,
                                                 const float* __restrict__ eob,
                                                 float* __restrict__ out)
{
    int tid   = threadIdx.x;
    int token = blockIdx.x;                      // g*S + s
    int g     = token / S;
    const float* cwr = cw + (size_t)token * E;
    float w[E];
#pragma unroll
    for (int e = 0; e < E; ++e) w[e] = 2.f * cwr[e];
    const float* eg = eob + (size_t)g * E * MDIM;
    float* orow = out + (size_t)token * MDIM;
    for (int m = tid; m < MDIM; m += 256) {
        float a = 0.f;
#pragma unroll
        for (int e = 0; e < E; ++e) a += w[e] * eg[(size_t)e * MDIM + m];
        orow[m] = a;
    }
}

// ---------------------------------------------------------------------------
extern "C" void kernel_launch(void* const* d_in, const int* in_sizes, int n_in,
                              void* d_out, int out_size, void* d_ws, size_t ws_size,
                              hipStream_t stream)
{
    const float* x  = (const float*)d_in[0];   // [G,S,M]
    const float* Wg = (const float*)d_in[1];   // [M,E]
    const float* bg = (const float*)d_in[2];   // [E]
    const float* wi = (const float*)d_in[3];   // [E,H,M]
    const float* wo = (const float*)d_in[4];   // [E,M,H]
    float* out = (float*)d_out;                // [G,S,M]

    // Workspace carve-up (~13.5 MB total), 256B aligned by construction.
    char* ws = (char*)d_ws;
    size_t off = 0;
    float* cw   = (float*)(ws + off); off += (size_t)G * S * E * 4;            // 512 KB
    float* dvp  = (float*)(ws + off); off += (size_t)SC * E * G * MDIM * 4;    // 8 MB
    float* dvec = (float*)(ws + off); off += (size_t)E * G * MDIM * 4;         // 512 KB
    float* hbuf = (float*)(ws + off); off += (size_t)E * H * 16 * 4;           // 4 MB
    float* eob  = (float*)(ws + off); off += (size_t)G * E * MDIM * 4;         // 512 KB
    (void)ws_size; (void)n_in; (void)in_sizes; (void)out_size;

    k_gating        <<<(G * S) / 8,               256, 0, stream>>>(x, Wg, bg, cw);
    k_dispatch_part <<<G * (MDIM / 256) * SC,     256, 0, stream>>>(x, cw, dvp);
    k_dispatch_reduce<<<(E * G * MDIM) / 256,     256, 0, stream>>>(dvp, dvec);
    k_expert_in     <<<(E * (H / 16)) / 8,        256, 0, stream>>>(wi, dvec, hbuf);
    k_expert_out    <<<(E * (MDIM / 16)) / 8,     256, 0, stream>>>(wo, hbuf, eob);
    k_combine       <<<G * S,                     256, 0, stream>>>(cw, eob, out);
}